// VQVAE_71554155151956
// MI455X (gfx1250) — compile-verified
//
#include <hip/hip_runtime.h>

// ---------------- problem constants ----------------
#define N_ROWS   131072      // 32*64*64
#define DIM      64
#define NE       1024
#define DECAY    0.99f
#define OMD      0.01f       // 1 - DECAY
#define EPSV     1e-5f

#define WAVES        8
#define ROWS_PER_WG  128     // 8 waves * 16 rows
#define CHUNK        128     // codes staged in LDS per pass
#define CT           8       // 16-code tiles per chunk
#define NCHUNK       (NE / CHUNK)

typedef float v2f __attribute__((ext_vector_type(2)));
typedef float v4f __attribute__((ext_vector_type(4)));
typedef float v8f __attribute__((ext_vector_type(8)));

// ---------------- workspace layout (float index) ----------------
#define WS_DIFF    0
#define WS_ONEHOT  256
#define WS_ESUM    2048
#define WS_E2      (2048 + 65536)
#define WS_ZERO_FLOATS (2048 + 65536)   // zero diff+onehot+esum each launch

// ---------------- output layout (float index) ----------------
#define OUT_Q     0
#define OUT_DIFF  8388608
#define OUT_IND   8388609
#define OUT_NCS   8519681
#define OUT_NEA   8520705
#define OUT_NEMB  8586241

// ||e_c||^2 per code
__global__ __launch_bounds__(128) void vq_e2_kernel(const float* __restrict__ embed,
                                                    float* __restrict__ e2) {
  int c = blockIdx.x * 128 + threadIdx.x;
  if (c < NE) {
    float s = 0.f;
    #pragma unroll
    for (int k = 0; k < DIM; ++k) { float v = embed[k * NE + c]; s += v * v; }
    e2[c] = s;
  }
}

__global__ __launch_bounds__(256) void vq_main_kernel(const float* __restrict__ x,
                                                      const float* __restrict__ embed,
                                                      const float* __restrict__ e2,
                                                      float* __restrict__ out,
                                                      float* __restrict__ ws) {
  __shared__ float ldsB[CT * 16 * 32 * 2];   // 32 KB: B fragments (t, step, lane, 2)
  __shared__ float ldsE2[CHUNK];
  __shared__ int   idxSh[WAVES][16];

  const int tid  = threadIdx.x;
  const int wave = tid >> 5;
  const int lane = tid & 31;
  const int lh   = lane >> 4;        // which 16-lane half
  const int lm   = lane & 15;
  const int rowBase = blockIdx.x * ROWS_PER_WG + wave * 16;

  // --- A fragments: 16 rows x K=64, held in registers for the whole kernel ---
  // ISA 32-bit A 16x4 layout: lanes 0-15 -> K={4s,4s+1}; lanes 16-31 -> K={4s+2,4s+3}
  v2f aFrag[16];
  const float* xrow = x + (size_t)(rowBase + lm) * DIM + 2 * lh;
  #pragma unroll
  for (int s = 0; s < 16; ++s) aFrag[s] = *(const v2f*)(xrow + 4 * s);

  float best[8];
  int   bidx[8];
  #pragma unroll
  for (int r = 0; r < 8; ++r) { best[r] = 3.4e38f; bidx[r] = 0; }

  for (int ch = 0; ch < NCHUNK; ++ch) {
    const int cbase = ch * CHUNK;
    __syncthreads();   // previous chunk fully consumed
    // stage embed[0:64, cbase:cbase+128] into WMMA B-fragment order (float4 reads)
    #pragma unroll
    for (int it = 0; it < (CHUNK * DIM) / (256 * 4); ++it) {
      int e  = (it * 256 + tid) * 4;
      int k  = e >> 7;                 // 0..63
      int cc = e & (CHUNK - 1);        // 0..124, multiple of 4 (stays in one 16-code tile)
      v4f v  = *(const v4f*)(embed + (size_t)k * NE + cbase + cc);
      int t  = cc >> 4;
      int s  = k >> 2;
      int j  = k & 1;
      int lp = ((k & 3) >> 1) * 16 + (cc & 15);
      #pragma unroll
      for (int u = 0; u < 4; ++u)
        ldsB[((t * 16 + s) * 32 + lp + u) * 2 + j] = v[u];
    }
    if (tid < CHUNK) ldsE2[tid] = e2[cbase + tid];
    // prefetch next chunk's embed rows (64 rows x 512B -> 512 x 64B lines)
    if (ch + 1 < NCHUNK) {
      #pragma unroll
      for (int u = 0; u < 2; ++u) {
        int l  = tid + u * 256;
        int kk = l >> 3;
        const char* p = (const char*)(embed + (size_t)kk * NE + cbase + CHUNK) + (l & 7) * 64;
        __builtin_prefetch(p, 0, 0);
      }
    }
    __syncthreads();

    // double-buffered B fragments in registers; DS latency hidden behind WMMA chain
    v2f bf[2][16];
    #pragma unroll
    for (int s = 0; s < 16; ++s)
      bf[0][s] = *(const v2f*)&ldsB[((0 * 16 + s) * 32 + lane) * 2];

    #pragma unroll
    for (int t = 0; t < CT; ++t) {
      if (t + 1 < CT) {
        #pragma unroll
        for (int s = 0; s < 16; ++s)
          bf[(t + 1) & 1][s] = *(const v2f*)&ldsB[(((t + 1) * 16 + s) * 32 + lane) * 2];
      }
      v8f acc = {0.f, 0.f, 0.f, 0.f, 0.f, 0.f, 0.f, 0.f};
      #pragma unroll
      for (int s = 0; s < 16; ++s) {
        acc = __builtin_amdgcn_wmma_f32_16x16x4_f32(
            /*neg_a=*/false, aFrag[s], /*neg_b=*/false, bf[t & 1][s],
            /*c_mod=*/(short)0, acc, /*reuse_a=*/false, /*reuse_b=*/false);
      }
      // score = ||e||^2 - 2 f.e   (||f||^2 is row-constant; argmin unchanged)
      float se2 = ldsE2[t * 16 + lm];
      int   code = cbase + t * 16 + lm;
      #pragma unroll
      for (int r = 0; r < 8; ++r) {
        float sc = se2 - 2.f * acc[r];
        if (sc < best[r]) { best[r] = sc; bidx[r] = code; }  // codes ascend -> first-min kept
      }
    }
  }

  // cross-lane argmin within each 16-lane half (rows 0-7 / 8-15)
  #pragma unroll
  for (int m = 1; m <= 8; m <<= 1) {
    #pragma unroll
    for (int r = 0; r < 8; ++r) {
      float ov = __shfl_xor(best[r], m, 32);
      int   oi = __shfl_xor(bidx[r], m, 32);
      if (ov < best[r] || (ov == best[r] && oi < bidx[r])) { best[r] = ov; bidx[r] = oi; }
    }
  }
  if (lm == 0) {
    #pragma unroll
    for (int r = 0; r < 8; ++r) idxSh[wave][lh * 8 + r] = bidx[r];
  }
  __syncthreads();

  // gather / straight-through / loss / segment-sum scatter
  float* qout = out + OUT_Q;
  float* esum = ws + WS_ESUM;
  float dacc = 0.f;
  for (int m = 0; m < 16; ++m) {
    int c = idxSh[wave][m];
    size_t ro = (size_t)(rowBase + m) * DIM;
    float xv0 = x[ro + lane];
    float xv1 = x[ro + 32 + lane];
    float q0 = embed[lane * NE + c];
    float q1 = embed[(lane + 32) * NE + c];
    qout[ro + lane]      = xv0 + (q0 - xv0);   // straight-through == quantize numerically
    qout[ro + 32 + lane] = xv1 + (q1 - xv1);
    float d0 = q0 - xv0, d1 = q1 - xv1;
    dacc += d0 * d0 + d1 * d1;
    atomicAdd(&esum[lane * NE + c], xv0);
    atomicAdd(&esum[(lane + 32) * NE + c], xv1);
    if (lane == 0) atomicAdd(&ws[WS_ONEHOT + c], 1.0f);
  }
  if (lane < 16) out[OUT_IND + rowBase + lane] = (float)idxSh[wave][lane];

  #pragma unroll
  for (int m = 1; m < 32; m <<= 1) dacc += __shfl_xor(dacc, m, 32);
  if (lane == 0) atomicAdd(&ws[WS_DIFF], dacc);
}

__global__ __launch_bounds__(1024) void vq_finalize_kernel(const float* __restrict__ cluster_size,
                                                           const float* __restrict__ embed_avg,
                                                           const float* __restrict__ ws,
                                                           float* __restrict__ out) {
  __shared__ float red[1024];
  int c = threadIdx.x;
  float ncs = DECAY * cluster_size[c] + OMD * ws[WS_ONEHOT + c];
  out[OUT_NCS + c] = ncs;
  red[c] = ncs;
  __syncthreads();
  for (int s = 512; s > 0; s >>= 1) {
    if (c < s) red[c] += red[c + s];
    __syncthreads();
  }
  float n  = red[0];
  float cs = (ncs + EPSV) / (n + (float)NE * EPSV) * n;
  #pragma unroll 4
  for (int k = 0; k < DIM; ++k) {
    float ea = DECAY * embed_avg[k * NE + c] + OMD * ws[WS_ESUM + k * NE + c];
    out[OUT_NEA + k * NE + c]  = ea;
    out[OUT_NEMB + k * NE + c] = ea / cs;
  }
  if (c == 0) out[OUT_DIFF] = ws[WS_DIFF] * (1.0f / 8388608.0f);
}

extern "C" void kernel_launch(void* const* d_in, const int* in_sizes, int n_in,
                              void* d_out, int out_size, void* d_ws, size_t ws_size,
                              hipStream_t stream) {
  (void)in_sizes; (void)n_in; (void)out_size; (void)ws_size;
  const float* x            = (const float*)d_in[0];
  const float* embed        = (const float*)d_in[1];
  const float* cluster_size = (const float*)d_in[2];
  const float* embed_avg    = (const float*)d_in[3];
  float* out = (float*)d_out;
  float* ws  = (float*)d_ws;

  hipMemsetAsync(d_ws, 0, (size_t)WS_ZERO_FLOATS * sizeof(float), stream);
  vq_e2_kernel<<<NE / 128, 128, 0, stream>>>(embed, ws + WS_E2);
  vq_main_kernel<<<N_ROWS / ROWS_PER_WG, 256, 0, stream>>>(x, embed, ws + WS_E2, out, ws);
  vq_finalize_kernel<<<1, 1024, 0, stream>>>(cluster_size, embed_avg, ws, out);
}